// LateralInhibitionLIFCell_55740085567939
// MI455X (gfx1250) — compile-verified
//
#include <hip/hip_runtime.h>
#include <hip/hip_bf16.h>

// ---------------------------------------------------------------------------
// AdEx-LIF elementwise step + winner-take-all on batch row 0.
// Memory-bound: 512 MB streamed @ 23.3 TB/s => ~22 us floor. All global
// traffic uses 128-bit vector accesses (global_load_b128/global_store_b128)
// with non-temporal cache hints (working set >> 192 MB L2, zero reuse).
// ---------------------------------------------------------------------------

typedef float v4f __attribute__((ext_vector_type(4)));

#define V_TH       1.0f
#define V_PEAK     30.0f
#define V_RESET    0.0f
#define INHIBITION (-5.0f)

// Zero the 8-byte packed WTA key in workspace.
__global__ void lif_init_kernel(unsigned long long* __restrict__ key) {
    if (threadIdx.x == 0 && blockIdx.x == 0) *key = 0ull;
}

// Main elementwise pass. One thread = one 4-float chunk (16 bytes per tensor).
__global__ __launch_bounds__(256)
void lif_main_kernel(const float* __restrict__ x, const float* __restrict__ v,
                     const float* __restrict__ i_in, const float* __restrict__ w,
                     float* __restrict__ z_out, float* __restrict__ v_out,
                     float* __restrict__ i_out, float* __restrict__ w_out,
                     unsigned long long* __restrict__ key,
                     unsigned n_row0, unsigned total) {
    unsigned t    = blockIdx.x * blockDim.x + threadIdx.x;
    unsigned base = t * 4u;
    if (base >= total) return;

    const v4f xv = __builtin_nontemporal_load((const v4f*)(x    + base));
    const v4f vv = __builtin_nontemporal_load((const v4f*)(v    + base));
    const v4f iv = __builtin_nontemporal_load((const v4f*)(i_in + base));
    const v4f wv = __builtin_nontemporal_load((const v4f*)(w    + base));

    const bool row0 = (base < n_row0);

    v4f zr, vr, ir, wr;

#pragma unroll
    for (int c = 0; c < 4; ++c) {
        float xc = xv[c], vc = vv[c], ic = iv[c], wc = wv[c];

        // i_new = i + 0.5*(x - i)
        float inew = ic + 0.5f * (xc - ic);
        // exp_term = exp(v - V_TH)  (DELTA_T = 1)
        float eterm = expf(vc - V_TH);
        // v_new = v + 0.5*(-v + exp_term + i_new - w)   (V_REST = 0)
        float vnew = vc + 0.5f * (-vc + eterm + inew - wc);
        // w_new = w + 0.5*(0 - w)   (A_PARAM = 0)
        float wnew = 0.5f * wc;

        float zc = (vnew >= V_PEAK) ? 1.0f : 0.0f;
        float vbefore = vnew;
        if (zc != 0.0f) vnew = V_RESET;   // B_PARAM = 0 -> w unchanged on spike

        zr[c] = zc; vr[c] = vnew; ir[c] = inew; wr[c] = wnew;

        // Row-0 WTA candidate: vbefore >= 30 > 0, so IEEE bits are
        // order-preserving. Inverted index -> lowest index wins ties
        // (matches jnp.argmax). atomicMax is order-independent.
        if (row0 && zc != 0.0f) {
            unsigned j = base + (unsigned)c;
            unsigned long long k =
                ((unsigned long long)__float_as_uint(vbefore) << 32) |
                (unsigned long long)(0xFFFFFFFFu - j);
            atomicMax(key, k);
        }
    }

    __builtin_nontemporal_store(zr, (v4f*)(z_out + base));
    __builtin_nontemporal_store(vr, (v4f*)(v_out + base));
    __builtin_nontemporal_store(ir, (v4f*)(i_out + base));
    __builtin_nontemporal_store(wr, (v4f*)(w_out + base));
}

// Apply lateral inhibition to batch row 0 (runs after main kernel on stream).
__global__ __launch_bounds__(256)
void lif_fixup_kernel(float* __restrict__ v_row0, float* __restrict__ w_row0,
                      const unsigned long long* __restrict__ key, unsigned N) {
    unsigned long long k = *key;
    if (k == 0ull) return;                 // no spike in row 0 -> nothing to do
    unsigned winner = 0xFFFFFFFFu - (unsigned)(k & 0xFFFFFFFFull);
    unsigned j = blockIdx.x * blockDim.x + threadIdx.x;
    if (j < N && j != winner) {
        v_row0[j] = INHIBITION;
        w_row0[j] = 0.0f;
    }
}

extern "C" void kernel_launch(void* const* d_in, const int* in_sizes, int n_in,
                              void* d_out, int out_size, void* d_ws, size_t ws_size,
                              hipStream_t stream) {
    const float* x = (const float*)d_in[0];
    const float* v = (const float*)d_in[1];
    const float* i = (const float*)d_in[2];
    const float* w = (const float*)d_in[3];

    const unsigned total = (unsigned)in_sizes[0];   // B*N elements per tensor
    const unsigned B     = 256u;                    // fixed by reference
    const unsigned N     = total / B;               // row length (row 0 = WTA row)

    float* z_out = (float*)d_out;                   // outputs concatenated:
    float* v_out = z_out + (size_t)total;           // z, v_new, i_new, w_new
    float* i_out = z_out + 2ull * total;
    float* w_out = z_out + 3ull * total;

    unsigned long long* key = (unsigned long long*)d_ws;

    lif_init_kernel<<<1, 1, 0, stream>>>(key);

    unsigned chunks = total / 4u;                   // total is a multiple of 4
    unsigned blocks = (chunks + 255u) / 256u;
    lif_main_kernel<<<blocks, 256, 0, stream>>>(x, v, i, w,
                                                z_out, v_out, i_out, w_out,
                                                key, N, total);

    unsigned fix_blocks = (N + 255u) / 256u;
    lif_fixup_kernel<<<fix_blocks, 256, 0, stream>>>(v_out, w_out, key, N);
}